// super_voxels_analyze_23519240913464
// MI455X (gfx1250) — compile-verified
//
#include <hip/hip_runtime.h>
#include <math.h>

// ---------------------------------------------------------------------------
// super_voxels_analyze for MI455X (gfx1250, wave32)
//   cos einsum -> V_WMMA_F32_16X16X4_F32, with (cos+1) and the exp2 scale
//   folded into the WMMA operands:
//     A-rows = unit_a * cbrt(log2e),  C = cbrt(log2e)  =>  D = K3*(cos+1)
//     softmax exp: e = exp2(D^3)  (== exp((cos+1)^3))
//   padding rows (p>=24): A-row = 0, C = -64 => D^3 = -262144 => exp2 -> 0
// ---------------------------------------------------------------------------

typedef __attribute__((ext_vector_type(2))) float v2f;
typedef __attribute__((ext_vector_type(8))) float v8f;

#define SV_EPS        1e-8f
#define SV_CBRT_LOG2E 1.1299472763373901f   /* cbrt(log2(e)) */
#define SV_K2LOG90    12.983706192659348f   /* 2*log2(90) = 2*ln(90)*log2(e) */
#define SV_PAD_C      (-64.0f)              /* sentinel: exp2((-64)^3) == 0 */
#define SPLITV   8
#define WAVES_PER_BLOCK 8

__device__ __forceinline__ float fast_rcp(float x)  { return __builtin_amdgcn_rcpf(x); }
__device__ __forceinline__ float fast_sqrt(float x) { return __builtin_amdgcn_sqrtf(x); }

// ---------------------------------------------------------------------------
// Phase 1: per-voxel precompute (runs once; precise math).
//  Abuf : per (v, tile t in {0,1}, lane L) a v2f holding the WMMA A-operand
//         image of (unit_a * K3) rows (M = p = 16*t + L%16):
//           lanes 0..15  -> (K0,K1) = (K3*ux, K3*uy)
//           lanes 16..31 -> (K2,K3) = (K3*uz, 0)
//         p >= 24 -> zeros (padding rows, killed by the C sentinel).
//  anorm: a_norm[v][p] zero-padded to 32 entries per v.
// ---------------------------------------------------------------------------
__global__ void sv_prep(const float* __restrict__ sp,
                        float* __restrict__ Abuf,
                        float* __restrict__ anorm,
                        int V) {
    int idx = blockIdx.x * blockDim.x + threadIdx.x;
    if (idx >= V * 64) return;
    int v = idx >> 6;
    int r = idx & 63;
    int t = r >> 5;        // tile
    int L = r & 31;        // lane image
    int col = L & 15;
    int p = t * 16 + col;  // 0..31

    float ux = 0.f, uy = 0.f, uz = 0.f, n = 0.f;
    if (p < 24) {
        const float* a = sp + (size_t)v * 75 + (size_t)(1 + p) * 3;
        float x = a[0], y = a[1], z = a[2];
        n = sqrtf(fmaf(x, x, fmaf(y, y, z * z)));
        float inv = SV_CBRT_LOG2E / fmaxf(n, SV_EPS);   // fold K3 into unit_a
        ux = x * inv; uy = y * inv; uz = z * inv;
    }
    v2f e;
    if (L < 16) { e.x = ux; e.y = uy; }
    else        { e.x = uz; e.y = 0.f; }
    ((v2f*)Abuf)[(size_t)(v * 2 + t) * 32 + L] = e;
    if (L < 16) anorm[(size_t)v * 32 + p] = n;
}

// ---------------------------------------------------------------------------
// Phase 2: main compute. One block = one 16-query tile x one V-split chunk.
// Each wave handles voxels vi = (sv*8 + w), stride 64. Per v: two WMMAs give
// D = K3*(cos+1) for 16 queries x 24 p; lane L holds 8 consecutive p's for
// query L%16, so the softmax reduce is in-register adds + one shfl_xor(16).
// ---------------------------------------------------------------------------
__global__ void __launch_bounds__(256)
sv_main(const float* __restrict__ q,
        const float* __restrict__ sp,
        const float* __restrict__ val,
        const float* __restrict__ Abuf,
        const float* __restrict__ anorm,
        float* __restrict__ partial,
        int N, int V) {
    int b  = blockIdx.x;
    int qt = b / SPLITV;
    int sv = b % SPLITV;
    int tid = threadIdx.x;
    int w = tid >> 5;          // wave in block
    int L = tid & 31;          // lane
    int col = L & 15;
    bool hi = (L >= 16);

    int qidx = qt * 16 + col;
    if (qidx >= N) qidx = N - 1;           // harmless duplicate work
    float qx = q[(size_t)qidx * 3 + 0];
    float qy = q[(size_t)qidx * 3 + 1];
    float qz = q[(size_t)qidx * 3 + 2];

    const v2f* Ab = (const v2f*)Abuf;
    int base = hi ? 8 : 0;                 // first p held by this lane (tile 0)

    v8f c0, c1;                            // WMMA accumulators (constants)
#pragma unroll
    for (int j = 0; j < 8; ++j) {
        c0[j] = SV_CBRT_LOG2E;                        // all tile-0 rows valid
        c1[j] = hi ? SV_PAD_C : SV_CBRT_LOG2E;        // tile-1 hi rows = padding
    }

    float acc = 0.f;
    for (int vi = sv * WAVES_PER_BLOCK + w; vi < V; vi += SPLITV * WAVES_PER_BLOCK) {
        const float* c = sp + (size_t)vi * 75;
        float dx = qx - c[0];
        float dy = qy - c[1];
        float dz = qz - c[2];
        float dist = fast_sqrt(fmaf(dx, dx, fmaf(dy, dy, dz * dz)));
        float inv  = fast_rcp(fmaxf(dist, SV_EPS));

        // B operand: K x N (components x queries), zero-padded K=3 -> 4
        v2f B;
        B.x = (hi ? dz : dx) * inv;        // lanes<16: K0=ux ; lanes>=16: K2=uz
        B.y = hi ? 0.f : dy * inv;         // lanes<16: K1=uy ; lanes>=16: K3=0

        v2f A0 = Ab[(size_t)(vi * 2 + 0) * 32 + L];
        v2f A1 = Ab[(size_t)(vi * 2 + 1) * 32 + L];

        const float* nb = anorm + (size_t)vi * 32 + base;
        v8f na0 = *(const v8f*)(nb);        // a_norm for p = base..base+7
        v8f na1 = *(const v8f*)(nb + 16);   // a_norm for p = 16+base..16+base+7

        // D[j] @ lane L = K3*(cos+1) for p = 16*t + base + j, query = L%16
        v8f d0 = __builtin_amdgcn_wmma_f32_16x16x4_f32(
                     false, A0, false, B, (short)0, c0, false, false);
        v8f d1 = __builtin_amdgcn_wmma_f32_16x16x4_f32(
                     false, A1, false, B, (short)0, c1, false, false);

        float se = 0.f, sa = 0.f;
#pragma unroll
        for (int j = 0; j < 8; ++j) {
            float t1 = d0[j];
            float e  = __builtin_amdgcn_exp2f(t1 * t1 * t1);  // exp((cos+1)^3)
            se += e;
            sa  = fmaf(na0[j], e, sa);
        }
#pragma unroll
        for (int j = 0; j < 8; ++j) {
            float t1 = d1[j];                                  // padding -> -64
            float e  = __builtin_amdgcn_exp2f(t1 * t1 * t1);   // padding -> 0
            se += e;
            sa  = fmaf(na1[j], e, sa);
        }
        // combine the two lane-halves (each holds half the p-range per query)
        se += __shfl_xor(se, 16, 32);
        sa += __shfl_xor(sa, 16, 32);

        float corr = sa * fast_rcp(se);
        // sigmoid(2*ln(90)*(corr-dist)) = 1/(1+2^(-2*log2(90)*(corr-dist)))
        float e2   = __builtin_amdgcn_exp2f(SV_K2LOG90 * (dist - corr));
        float sig  = fast_rcp(1.0f + e2);   // saturates to 0/1 correctly
        acc = fmaf(val[vi], sig, acc);
    }

    // fixed-order in-block reduction over the 8 waves
    __shared__ float red[WAVES_PER_BLOCK * 16];
    if (!hi) red[w * 16 + col] = acc;   // halves are duplicates; keep low half
    __syncthreads();
    if (tid < 16) {
        float s = 0.f;
#pragma unroll
        for (int ww = 0; ww < WAVES_PER_BLOCK; ++ww) s += red[ww * 16 + tid];
        partial[(size_t)(qt * SPLITV + sv) * 16 + tid] = s;
    }
}

// ---------------------------------------------------------------------------
// Phase 3: fixed-order combine of the SPLITV partial sums + output predicate.
// ---------------------------------------------------------------------------
__global__ void sv_finish(const float* __restrict__ q,
                          const float* __restrict__ partial,
                          float* __restrict__ out,
                          int N) {
    int n = blockIdx.x * blockDim.x + threadIdx.x;
    if (n >= N) return;
    int qt = n >> 4;
    int m  = n & 15;
    float s = 0.f;
#pragma unroll
    for (int svv = 0; svv < SPLITV; ++svv)
        s += partial[(size_t)(qt * SPLITV + svv) * 16 + m];
    out[n] = (q[(size_t)n * 3] > -1.0f) ? s : 0.0f;
}

// ---------------------------------------------------------------------------
extern "C" void kernel_launch(void* const* d_in, const int* in_sizes, int n_in,
                              void* d_out, int out_size, void* d_ws, size_t ws_size,
                              hipStream_t stream) {
    const float* q   = (const float*)d_in[0];   // (N,3)
    const float* sp  = (const float*)d_in[1];   // (V,25,3)
    const float* val = (const float*)d_in[2];   // (V,)
    float* out = (float*)d_out;

    int N = in_sizes[0] / 3;
    int V = in_sizes[1] / 75;
    int QT = (N + 15) / 16;

    // workspace layout (floats): Abuf[V*128] | anorm[V*32] | partial[QT*SPLITV*16]
    float* Abuf    = (float*)d_ws;
    float* anorm   = Abuf + (size_t)V * 128;
    float* partial = anorm + (size_t)V * 32;

    {
        int total = V * 64;
        sv_prep<<<(total + 255) / 256, 256, 0, stream>>>(sp, Abuf, anorm, V);
    }
    sv_main<<<QT * SPLITV, 256, 0, stream>>>(q, sp, val, Abuf, anorm, partial, N, V);
    sv_finish<<<(N + 255) / 256, 256, 0, stream>>>(q, partial, out, N);
}